// GNNNet_2130303779216
// MI455X (gfx1250) — compile-verified
//
#include <hip/hip_runtime.h>
#include <hip/hip_bf16.h>
#include <cstdint>

#define NN 100000
#define NE 1600000

typedef float v2f __attribute__((ext_vector_type(2)));
typedef float v8f __attribute__((ext_vector_type(8)));

// ---- order-preserving float <-> uint encoding for atomic segment-max ----
__device__ __forceinline__ unsigned fenc(float x) {
    unsigned b = __float_as_uint(x);
    return (b & 0x80000000u) ? ~b : (b | 0x80000000u);
}
__device__ __forceinline__ float fdec(unsigned k) {
    unsigned b = (k & 0x80000000u) ? (k & 0x7FFFFFFFu) : ~k;
    return __uint_as_float(b);
}

// ============================================================================
// WMMA f32 GEMM: C[N, M] = A[N, K] @ B[K, M] (+bias, optional tanh epilogue)
//   M = TM*16 (compile-time).  One wave32 computes a full 16-row x M strip:
//   a single A-fragment (float2/lane) feeds TM chained V_WMMA_F32_16X16X4_F32
//   ops per k-step, so A is read exactly once from global.  The whole B panel
//   (K*M floats <= 32 KB) is staged in LDS by the workgroup, so B global
//   traffic is one pass per block instead of per wave per k-step.
// A-frag (16x4 f32): lanes 0-15 rows 0..15 K={0,1}; lanes 16-31 K={2,3}.
// B-frag (4x16 f32): lanes 0-15 K={0,1}; lanes 16-31 K={2,3}.
// C tile (16x16 f32, 8 VGPRs): vgpr i -> lanes 0-15 row i, lanes 16-31 row i+8.
// ============================================================================
template <int ACT, int TM>   // ACT: 0 = none, 2 = tanh;  M = TM*16
__global__ void wmma_gemm_f32(const float* __restrict__ A,
                              const float* __restrict__ B,
                              const float* __restrict__ bias,
                              float* __restrict__ C,
                              int N, int K) {
    constexpr int M = TM * 16;
    extern __shared__ float sB[];                 // [K, M]

    // cooperative B panel load (float4 vectorized)
    for (int i = threadIdx.x * 4; i < K * M; i += blockDim.x * 4)
        *(float4*)(sB + i) = *(const float4*)(B + i);
    __syncthreads();

    const int lane = threadIdx.x & 31;
    const int waveInBlk = threadIdx.x >> 5;
    const int wavesPerBlk = blockDim.x >> 5;
    const int strip = blockIdx.x * wavesPerBlk + waveInBlk;   // 16-row strip id
    if (strip * 16 >= N) return;
    const int r0 = strip << 4;
    const int half = lane >> 4;   // 0: K pair {0,1}, 1: K pair {2,3}
    const int l = lane & 15;

    v8f acc[TM];
#pragma unroll
    for (int t = 0; t < TM; ++t) acc[t] = (v8f){};

    const float* Arow = A + (size_t)(r0 + l) * K + (half << 1);
    __builtin_prefetch(Arow, 0, 0);               // global_prefetch_b8
    __builtin_prefetch(Arow + 64, 0, 0);

#pragma unroll 2
    for (int k = 0; k < K; k += 4) {
        const v2f a = *(const v2f*)(Arow + k);    // contiguous pair, 8B aligned
        const float* brow = sB + (k + (half << 1)) * M + l;
#pragma unroll
        for (int t = 0; t < TM; ++t) {
            v2f b;
            b.x = brow[t * 16];                   // row k+2*half
            b.y = brow[t * 16 + M];               // row k+2*half+1
            acc[t] = __builtin_amdgcn_wmma_f32_16x16x4_f32(
                false, a, false, b, (short)0, acc[t], false, false);
        }
    }

#pragma unroll
    for (int t = 0; t < TM; ++t) {
        const int c = t * 16 + l;
        const float bv = bias ? bias[c] : 0.0f;
#pragma unroll
        for (int i = 0; i < 8; ++i) {
            float v = acc[t][i] + bv;
            if (ACT == 2) v = tanhf(v);
            C[(size_t)(r0 + i + (half << 3)) * M + c] = v;
        }
    }
}

// ---- per-node attention logits: alpha_s[n]=h[n].a_s, alpha_d[n]=h[n].a_d ----
__global__ void alpha_kernel(const float* __restrict__ H,
                             const float* __restrict__ a_s,
                             const float* __restrict__ a_d,
                             float* __restrict__ os, float* __restrict__ od,
                             int N, int D) {
    const int wave = (int)((blockIdx.x * (size_t)blockDim.x + threadIdx.x) >> 5);
    const int lane = threadIdx.x & 31;
    if (wave >= N) return;
    float ss = 0.f, sd = 0.f;
    const float* hr = H + (size_t)wave * D;
    for (int d = lane; d < D; d += 32) {
        const float h = hr[d];
        ss += h * a_s[d];
        sd += h * a_d[d];
    }
    for (int off = 16; off; off >>= 1) {
        ss += __shfl_xor(ss, off, 32);
        sd += __shfl_xor(sd, off, 32);
    }
    if (lane == 0) { os[wave] = ss; od[wave] = sd; }
}

__global__ void init_seg(unsigned* __restrict__ m, float* __restrict__ denom, int N) {
    const int i = blockIdx.x * blockDim.x + threadIdx.x;
    if (i < N) { m[i] = 0u; denom[i] = 0.f; }   // enc key 0 < enc(any real)
}

__global__ void zero_f(float* __restrict__ p, int n) {
    const int i = blockIdx.x * blockDim.x + threadIdx.x;
    if (i < n) p[i] = 0.f;
}

// pass 1: e = leaky_relu(a_s[src]+a_d[dst]); segment max via atomicMax
__global__ void edge_pass1(const long long* __restrict__ src,
                           const long long* __restrict__ dst,
                           const float* __restrict__ as, const float* __restrict__ ad,
                           float* __restrict__ ebuf, unsigned* __restrict__ m, int E) {
    const int i = blockIdx.x * blockDim.x + threadIdx.x;
    if (i >= E) return;
    const int s = (int)src[i], d = (int)dst[i];
    float e = as[s] + ad[d];
    e = e > 0.f ? e : 0.2f * e;
    ebuf[i] = e;
    atomicMax(&m[d], fenc(e));
}

// pass 2: w = exp(e - m[dst]); denom[dst] += w
__global__ void edge_pass2(const long long* __restrict__ dst,
                           float* __restrict__ ebuf, const unsigned* __restrict__ m,
                           float* __restrict__ denom, int E) {
    const int i = blockIdx.x * blockDim.x + threadIdx.x;
    if (i >= E) return;
    const int d = (int)dst[i];
    const float w = __expf(ebuf[i] - fdec(m[d]));
    ebuf[i] = w;
    atomicAdd(&denom[d], w);
}

// pass 3: ACC[dst] += H[src] * (w / (denom[dst]+eps)); float4 chunks
// lgc = log2(D/4): chunk index via shift/mask (no integer division)
__global__ void edge_pass3(const long long* __restrict__ src,
                           const long long* __restrict__ dst,
                           const float* __restrict__ ebuf,
                           const float* __restrict__ denom,
                           const float* __restrict__ H, float* __restrict__ ACC,
                           int E, int D, int lgc) {
    const int gid = blockIdx.x * blockDim.x + threadIdx.x;
    const int i = gid >> lgc;
    if (i >= E) return;
    const int c = gid & ((1 << lgc) - 1);
    const int s = (int)src[i], d = (int)dst[i];
    const float coef = ebuf[i] / (denom[d] + 1e-16f);
    const float4 h4 = *(const float4*)(H + (size_t)s * D + (c << 2));
    float* o = ACC + (size_t)d * D + (c << 2);
    atomicAdd(o + 0, h4.x * coef);
    atomicAdd(o + 1, h4.y * coef);
    atomicAdd(o + 2, h4.z * coef);
    atomicAdd(o + 3, h4.w * coef);
}

// G = relu(ACC + b);  D must be a power of two (64 / 128)
__global__ void bias_relu(const float* __restrict__ ACC, const float* __restrict__ b,
                          float* __restrict__ G, int total, int dmask) {
    const int i = blockIdx.x * blockDim.x + threadIdx.x;
    if (i >= total) return;
    G[i] = fmaxf(ACC[i] + b[i & dmask], 0.f);
}

__global__ void zero1(float* p) { *p = 0.f; }

// value = mean(z @ linV_w + linV_b); one wave per node, atomicAdd of /N share
__global__ void value_kernel(const float* __restrict__ Z,
                             const float* __restrict__ vw, const float* __restrict__ vb,
                             float* __restrict__ out, int N, int D) {
    const int wave = (int)((blockIdx.x * (size_t)blockDim.x + threadIdx.x) >> 5);
    const int lane = threadIdx.x & 31;
    if (wave >= N) return;
    float s = 0.f;
    const float* zr = Z + (size_t)wave * D;
    for (int d = lane; d < D; d += 32) s += zr[d] * vw[d];
    for (int off = 16; off; off >>= 1) s += __shfl_xor(s, off, 32);
    if (lane == 0) atomicAdd(out, (s + vb[0]) * (1.0f / (float)N));
}

static inline int cdiv(long long a, long long b) { return (int)((a + b - 1) / b); }

extern "C" void kernel_launch(void* const* d_in, const int* in_sizes, int n_in,
                              void* d_out, int out_size, void* d_ws, size_t ws_size,
                              hipStream_t stream) {
    const float*     x      = (const float*)d_in[0];
    const long long* ei     = (const long long*)d_in[1];
    const float*     W1     = (const float*)d_in[2];
    const float*     a_s1   = (const float*)d_in[3];
    const float*     a_d1   = (const float*)d_in[4];
    const float*     b1     = (const float*)d_in[5];
    const float*     W2     = (const float*)d_in[6];
    const float*     a_s2   = (const float*)d_in[7];
    const float*     a_d2   = (const float*)d_in[8];
    const float*     b2     = (const float*)d_in[9];
    const float*     lin1_w = (const float*)d_in[10];
    const float*     lin1_b = (const float*)d_in[11];
    const float*     lin2_w = (const float*)d_in[12];
    const float*     lin2_b = (const float*)d_in[13];
    const float*     l11_w  = (const float*)d_in[14];
    const float*     l11_b  = (const float*)d_in[15];
    const float*     lV_w   = (const float*)d_in[16];
    const float*     lV_b   = (const float*)d_in[17];

    const long long* src = ei;
    const long long* dst = ei + NE;

    // workspace layout (fp32 node tensors sized to fit the 192MB L2)
    float*    bufA  = (float*)d_ws;                       // [NN,128]
    float*    bufB  = bufA + (size_t)NN * 128;            // [NN,128]
    float*    as    = bufB + (size_t)NN * 128;            // [NN]
    float*    ad    = as + NN;                            // [NN]
    unsigned* mEnc  = (unsigned*)(ad + NN);               // [NN]
    float*    denom = (float*)(mEnc + NN);                // [NN]
    float*    ebuf  = denom + NN;                         // [NE]

    float* outF   = (float*)d_out;                        // [NN,64]
    float* valPtr = outF + (size_t)NN * 64;               // scalar

    const int TB = 256;
    const int stripBlocks = cdiv(NN / 16, TB / 32);       // one 16-row strip / wave

    // ---------- GAT layer 1 (128 -> 64) ----------
    wmma_gemm_f32<0, 4><<<stripBlocks, TB, 128 * 64 * 4, stream>>>(
        x, W1, nullptr, bufA, NN, 128);                           // H1 -> bufA
    alpha_kernel<<<cdiv((long long)NN * 32, TB), TB, 0, stream>>>(
        bufA, a_s1, a_d1, as, ad, NN, 64);
    init_seg<<<cdiv(NN, TB), TB, 0, stream>>>(mEnc, denom, NN);
    zero_f<<<cdiv((long long)NN * 64, TB), TB, 0, stream>>>(bufB, NN * 64);
    edge_pass1<<<cdiv(NE, TB), TB, 0, stream>>>(src, dst, as, ad, ebuf, mEnc, NE);
    edge_pass2<<<cdiv(NE, TB), TB, 0, stream>>>(dst, ebuf, mEnc, denom, NE);
    edge_pass3<<<cdiv((long long)NE * 16, TB), TB, 0, stream>>>(
        src, dst, ebuf, denom, bufA, bufB, NE, 64, 4);            // ACC1 -> bufB
    bias_relu<<<cdiv((long long)NN * 64, TB), TB, 0, stream>>>(
        bufB, b1, bufA, NN * 64, 63);                             // G1 -> bufA

    // ---------- GAT layer 2 (64 -> 128) ----------
    wmma_gemm_f32<0, 8><<<stripBlocks, TB, 64 * 128 * 4, stream>>>(
        bufA, W2, nullptr, bufB, NN, 64);                         // H2 -> bufB
    alpha_kernel<<<cdiv((long long)NN * 32, TB), TB, 0, stream>>>(
        bufB, a_s2, a_d2, as, ad, NN, 128);
    init_seg<<<cdiv(NN, TB), TB, 0, stream>>>(mEnc, denom, NN);
    zero_f<<<cdiv((long long)NN * 128, TB), TB, 0, stream>>>(bufA, NN * 128);
    edge_pass1<<<cdiv(NE, TB), TB, 0, stream>>>(src, dst, as, ad, ebuf, mEnc, NE);
    edge_pass2<<<cdiv(NE, TB), TB, 0, stream>>>(dst, ebuf, mEnc, denom, NE);
    edge_pass3<<<cdiv((long long)NE * 32, TB), TB, 0, stream>>>(
        src, dst, ebuf, denom, bufB, bufA, NE, 128, 5);           // ACC2 -> bufA
    bias_relu<<<cdiv((long long)NN * 128, TB), TB, 0, stream>>>(
        bufA, b2, bufB, NN * 128, 127);                           // G2 -> bufB

    // ---------- MLP head ----------
    wmma_gemm_f32<0, 4><<<stripBlocks, TB, 128 * 64 * 4, stream>>>(
        bufB, lin1_w, lin1_b, bufA, NN, 128);                     // T1 -> bufA
    wmma_gemm_f32<0, 4><<<stripBlocks, TB, 64 * 64 * 4, stream>>>(
        bufA, lin2_w, lin2_b, bufB, NN, 64);                      // Z  -> bufB
    wmma_gemm_f32<2, 4><<<stripBlocks, TB, 64 * 64 * 4, stream>>>(
        bufB, l11_w, l11_b, outF, NN, 64);                        // out = tanh(...)

    zero1<<<1, 1, 0, stream>>>(valPtr);
    value_kernel<<<cdiv((long long)NN * 32, TB), TB, 0, stream>>>(
        bufB, lV_w, lV_b, valPtr, NN, 64);
}